// NCNModuleCuda_20358144983216
// MI455X (gfx1250) — compile-verified
//
#include <hip/hip_runtime.h>

// Fused NCN module for MI455X (gfx1250, wave32).
//
// Memory-bound op: 537 MB min traffic @ 23.3 TB/s => ~23 us floor.
// Single fused kernel keeps x/xa in VGPRs across all 4 iterations.
// CDNA5 paths: global_load_async_to_lds_b128 (+ s_wait_asynccnt) stages W
// into LDS, ds_load reads it back, nontemporal b128 global loads/stores
// stream x/xa, wave32 shuffle butterflies do the 64-wide softmax reductions.

typedef float v4f __attribute__((ext_vector_type(4)));

#define BLOCK 256

__global__ __launch_bounds__(BLOCK) void ncn_fused_kernel(
    const float* __restrict__ x,
    const float* __restrict__ xa,
    const float* __restrict__ W,
    float* __restrict__ out,
    long long n)  // elements per tensor (8*4096*1024)
{
    // w1[1024] | w2[1024]; only shared object => LDS byte offset 0.
    __shared__ float Wlds[2048];

    const int tid = threadIdx.x;

    // ---- Stage W (8 KB) into LDS via CDNA5 async global->LDS DMA ----
    // Each lane copies 16B from each half of W: 8 waves x 32 lanes x 16B x 2 = 8 KB.
    {
        unsigned lds0 = (unsigned)(tid * 16);
        unsigned lds1 = lds0 + 4096u;
        const float* g0 = W + tid * 4;
        const float* g1 = W + 1024 + tid * 4;
        asm volatile("global_load_async_to_lds_b128 %0, %1, off"
                     :: "v"(lds0), "v"(g0) : "memory");
        asm volatile("global_load_async_to_lds_b128 %0, %1, off"
                     :: "v"(lds1), "v"(g1) : "memory");
        asm volatile("s_wait_asynccnt 0x0" ::: "memory");
    }
    __syncthreads();

    // 4 elements per thread; 16 lanes form one 64-wide softmax group.
    // Lanes 0-15 of a wave = one group, lanes 16-31 = the next group, so
    // xor-masks 1,2,4,8 never cross a group boundary (wave32).
    const long long t = (long long)blockIdx.x * BLOCK + tid;
    const long long base = t * 4;
    if (base >= n) return;

    const int dpos = (int)(base & 1023);  // position within embed dim D=1024

    // Direct __shared__ element access keeps these in addrspace(3) => ds_load.
    v4f w1 = { Wlds[dpos + 0], Wlds[dpos + 1], Wlds[dpos + 2], Wlds[dpos + 3] };
    v4f w2 = { Wlds[1024 + dpos + 0], Wlds[1024 + dpos + 1],
               Wlds[1024 + dpos + 2], Wlds[1024 + dpos + 3] };

    // Streamed exactly once: nontemporal to bypass L2 retention.
    v4f xv  = __builtin_nontemporal_load((const v4f*)(x  + base));
    v4f xav = __builtin_nontemporal_load((const v4f*)(xa + base));

    const float A   = 0.9f;
    const float OMA = (float)(1.0 - 0.9);  // matches JAX f32 rounding of (1.0-ALPHA)

#pragma unroll
    for (int it = 0; it < 4; ++it) {
        v4f pre = xv * w1 + xav * w2;

        // max over the 64-element group: local 4-way + 4 butterfly steps
        float m = fmaxf(fmaxf(pre.x, pre.y), fmaxf(pre.z, pre.w));
        m = fmaxf(m, __shfl_xor(m, 1, 32));
        m = fmaxf(m, __shfl_xor(m, 2, 32));
        m = fmaxf(m, __shfl_xor(m, 4, 32));
        m = fmaxf(m, __shfl_xor(m, 8, 32));

        v4f e;
        e.x = expf(pre.x - m);
        e.y = expf(pre.y - m);
        e.z = expf(pre.z - m);
        e.w = expf(pre.w - m);

        float s = (e.x + e.y) + (e.z + e.w);
        s += __shfl_xor(s, 1, 32);
        s += __shfl_xor(s, 2, 32);
        s += __shfl_xor(s, 4, 32);
        s += __shfl_xor(s, 8, 32);

        v4f y = e / s;                // softmax
        xav = xav * A + y * OMA;      // EMA update
        xv  = y;
    }

    __builtin_nontemporal_store(xv,  (v4f*)(out + base));
    __builtin_nontemporal_store(xav, (v4f*)(out + n + base));
}

extern "C" void kernel_launch(void* const* d_in, const int* in_sizes, int n_in,
                              void* d_out, int out_size, void* d_ws, size_t ws_size,
                              hipStream_t stream) {
    const float* x  = (const float*)d_in[0];
    const float* xa = (const float*)d_in[1];
    const float* W  = (const float*)d_in[2];
    float* out = (float*)d_out;

    long long n = (long long)in_sizes[0];          // 8*4096*1024 = 33,554,432
    long long threads = n / 4;                     // 4 elems per thread
    int blocks = (int)((threads + BLOCK - 1) / BLOCK);

    ncn_fused_kernel<<<blocks, BLOCK, 0, stream>>>(x, xa, W, out, n);
}